// Filter_long_16724602650774
// MI455X (gfx1250) — compile-verified
//
#include <hip/hip_runtime.h>
#include <cfloat>

typedef float v2f __attribute__((ext_vector_type(2)));
typedef float v8f __attribute__((ext_vector_type(8)));

#define L_DIM 10
#define C_DIM 256
#define D_DIM 64

// ---------------------------------------------------------------------------
// Kernel 1: precompute Mt[256][64] = (Wq @ (cluster@Wk)^T)^T * 1/sqrt(64)
//           and      V [256][64] = cluster @ Wv
// Tiny (~6 MFLOP total): plain VALU, one block per cluster row.
// ---------------------------------------------------------------------------
__global__ void flt_prep_kernel(const float* __restrict__ cluster,
                                const float* __restrict__ Wq,
                                const float* __restrict__ Wk,
                                const float* __restrict__ Wv,
                                float* __restrict__ Mt,
                                float* __restrict__ V) {
  __shared__ float cl[D_DIM];
  __shared__ float kr[D_DIM];
  const int c = blockIdx.x;       // 0..255
  const int t = threadIdx.x;      // 0..63
  cl[t] = cluster[c * D_DIM + t];
  __syncthreads();
  float accK = 0.f, accV = 0.f;
  #pragma unroll 4
  for (int f = 0; f < D_DIM; ++f) {
    const float cf = cl[f];
    accK += cf * Wk[f * D_DIM + t];
    accV += cf * Wv[f * D_DIM + t];
  }
  kr[t] = accK;
  V[c * D_DIM + t] = accV;
  __syncthreads();
  float accM = 0.f;
  #pragma unroll 4
  for (int e = 0; e < D_DIM; ++e) accM += kr[e] * Wq[t * D_DIM + e];
  Mt[c * D_DIM + t] = accM * 0.125f;   // fold 1/sqrt(D)
}

// ---------------------------------------------------------------------------
// Kernel 2: one wave32 per batch element.
//   A (16x64 fp32, rows 0..9 = short_l, 10..15 = 0) in registers
//   scores via V_WMMA_F32_16X16X4_F32 over 16 N-tiles x 16 k-steps
//   online softmax fused with uv-dot (no score storage)
//   new_user = sum_l s_l * w_filters[u,l,:]  ; plus pos/neg gathers
// ---------------------------------------------------------------------------
__global__ __launch_bounds__(256)
void flt_main_kernel(const int*   __restrict__ items,
                     const float* __restrict__ time_embs,
                     const float* __restrict__ user_emb,
                     const float* __restrict__ item_emb,
                     const float* __restrict__ w_filters,
                     const float* __restrict__ Mt,
                     const float* __restrict__ V,
                     const int*   __restrict__ users,
                     const int*   __restrict__ pos_items,
                     const int*   __restrict__ neg_items,
                     const float* __restrict__ a1p,
                     const float* __restrict__ a2p,
                     const float* __restrict__ a3p,
                     float* __restrict__ out, int B) {
  __shared__ float s_ue[8][D_DIM];
  __shared__ float s_uv[8][C_DIM];
  __shared__ float s_s [8][16];

  const int w    = threadIdx.x >> 5;
  const int lane = threadIdx.x & 31;
  int b = blockIdx.x * 8 + w;
  if (b >= B) b = B - 1;               // tail waves recompute last element (identical writes)
  const int u = users[b];

  // ---- user embedding -> LDS (and kept in regs for the final mix) ----
  const float2 ue2 = *(const float2*)(user_emb + u * D_DIM + 2 * lane);
  s_ue[w][2 * lane]     = ue2.x;
  s_ue[w][2 * lane + 1] = ue2.y;
  __syncthreads();

  // ---- uv[c] = user_emb . V[c]  (V is L2-hot, shared by all waves) ----
  #pragma unroll
  for (int j = 0; j < 8; ++j) {
    const int c = lane + 32 * j;
    const float4* vr = (const float4*)(V + c * D_DIM);
    float acc = 0.f;
    #pragma unroll
    for (int q = 0; q < 16; ++q) {
      const float4 vv = vr[q];
      acc += vv.x * s_ue[w][4 * q]     + vv.y * s_ue[w][4 * q + 1] +
             vv.z * s_ue[w][4 * q + 2] + vv.w * s_ue[w][4 * q + 3];
    }
    s_uv[w][c] = acc;
  }
  __syncthreads();

  // ---- build A (16x64) in registers, per documented 32-bit A layout ----
  // lane r = lane&15 holds row r; lane half selects K pair {khalf, khalf+1}
  const int r     = lane & 15;
  const int khalf = (lane >> 4) << 1;
  v2f A[16];
  if (r < L_DIM) {
    const int it = items[u * L_DIM + r];
    const float* itr = item_emb  + it * D_DIM;
    const float* ter = time_embs + (u * L_DIM + r) * D_DIM;
    #pragma unroll
    for (int t = 0; t < 16; ++t) {
      const int d0 = 4 * t + khalf;
      const float2 i2 = *(const float2*)(itr + d0);
      const float2 t2 = *(const float2*)(ter + d0);
      A[t].x = s_ue[w][d0]     + i2.x + t2.x;
      A[t].y = s_ue[w][d0 + 1] + i2.y + t2.y;
    }
  } else {
    #pragma unroll
    for (int t = 0; t < 16; ++t) { A[t].x = 0.f; A[t].y = 0.f; }
  }

  // ---- scores GEMM (WMMA fp32) + online softmax + uv-dot, tile by tile ----
  float m[8], Z[8], Wa[8];
  #pragma unroll
  for (int j = 0; j < 8; ++j) { m[j] = -FLT_MAX; Z[j] = 0.f; Wa[j] = 0.f; }

  const int n = lane & 15;
  for (int nt = 0; nt < 16; ++nt) {
    v8f cacc = {0.f, 0.f, 0.f, 0.f, 0.f, 0.f, 0.f, 0.f};
    const float* mtb = Mt + (nt * 16 + n) * D_DIM;   // Mt[c][k], c = nt*16+n
    #pragma unroll
    for (int t = 0; t < 16; ++t) {
      const int k0 = 4 * t + khalf;
      const float2 b2 = *(const float2*)(mtb + k0);
      v2f bb; bb.x = b2.x; bb.y = b2.y;
      cacc = __builtin_amdgcn_wmma_f32_16x16x4_f32(
          /*neg_a=*/false, A[t], /*neg_b=*/false, bb,
          /*c_mod=*/(short)0, cacc, /*reuse_a=*/false, /*reuse_b=*/false);
    }
    const float uvc = s_uv[w][nt * 16 + n];
    #pragma unroll
    for (int j = 0; j < 8; ++j) {
      float v = cacc[j];                      // row (j or j+8), col nt*16+n
      float tmax = v;
      #pragma unroll
      for (int mk = 1; mk < 16; mk <<= 1)
        tmax = fmaxf(tmax, __shfl_xor(tmax, mk, 32));
      const float nm = fmaxf(m[j], tmax);
      const float e  = __expf(v - nm);
      float ts = e, tw = e * uvc;
      #pragma unroll
      for (int mk = 1; mk < 16; mk <<= 1) {
        ts += __shfl_xor(ts, mk, 32);
        tw += __shfl_xor(tw, mk, 32);
      }
      const float sc = __expf(m[j] - nm);
      Z[j]  = Z[j]  * sc + ts;
      Wa[j] = Wa[j] * sc + tw;
      m[j]  = nm;
    }
  }

  // s_l = Wa/Z for row l (rows 10..15 are padding, unused)
  if (n == 0) {
    const int base = (lane >> 4) * 8;
    #pragma unroll
    for (int j = 0; j < 8; ++j) s_s[w][base + j] = Wa[j] / Z[j];
  }
  __syncthreads();

  // ---- new_user[e] = sum_l s_l * w_filters[u,l,e]; mix; gathers ----
  float acc0 = 0.f, acc1 = 0.f;
  const float* wfu = w_filters + u * L_DIM * D_DIM;
  #pragma unroll
  for (int l = 0; l < L_DIM; ++l) {
    const float sl = s_s[w][l];
    const float2 w2 = *(const float2*)(wfu + l * D_DIM + 2 * lane);
    acc0 += sl * w2.x;
    acc1 += sl * w2.y;
  }
  const float a1 = a1p[0], a2 = a2p[0], a3 = a3p[0];
  float2 og;
  og.x = a3 * (a1 * ue2.x + a2 * acc0);
  og.y = a3 * (a1 * ue2.y + a2 * acc1);
  *(float2*)(out + b * D_DIM + 2 * lane) = og;

  const int pi = pos_items[b], ni = neg_items[b];
  *(float2*)(out + (size_t)(B + b) * D_DIM + 2 * lane) =
      *(const float2*)(item_emb + pi * D_DIM + 2 * lane);
  *(float2*)(out + (size_t)(2 * B + b) * D_DIM + 2 * lane) =
      *(const float2*)(item_emb + ni * D_DIM + 2 * lane);
}

// ---------------------------------------------------------------------------
extern "C" void kernel_launch(void* const* d_in, const int* in_sizes, int n_in,
                              void* d_out, int out_size, void* d_ws, size_t ws_size,
                              hipStream_t stream) {
  const int*   items     = (const int*)  d_in[0];
  const float* time_embs = (const float*)d_in[1];
  const float* user_emb  = (const float*)d_in[2];
  const float* item_emb  = (const float*)d_in[3];
  const float* cluster   = (const float*)d_in[4];
  const float* w_filters = (const float*)d_in[5];
  const float* Wq        = (const float*)d_in[6];
  const float* Wk        = (const float*)d_in[7];
  const float* Wv        = (const float*)d_in[8];
  const int*   users     = (const int*)  d_in[9];
  const int*   pos_items = (const int*)  d_in[10];
  const int*   neg_items = (const int*)  d_in[11];
  const float* a1        = (const float*)d_in[12];
  const float* a2        = (const float*)d_in[13];
  const float* a3        = (const float*)d_in[14];

  const int B = in_sizes[9];

  float* Mt = (float*)d_ws;                 // [256][64]
  float* V  = Mt + C_DIM * D_DIM;           // [256][64]

  flt_prep_kernel<<<C_DIM, D_DIM, 0, stream>>>(cluster, Wq, Wk, Wv, Mt, V);

  const int blocks = (B + 7) / 8;           // 8 waves (batch elements) per block
  flt_main_kernel<<<blocks, 256, 0, stream>>>(
      items, time_embs, user_emb, item_emb, w_filters, Mt, V,
      users, pos_items, neg_items, a1, a2, a3, (float*)d_out, B);
}